// HeteroRGCNLayer_11716670783741
// MI455X (gfx1250) — compile-verified
//
#include <hip/hip_runtime.h>
#include <hip/hip_bf16.h>

typedef __attribute__((ext_vector_type(2))) float v2f;
typedef __attribute__((ext_vector_type(8))) float v8f;

#define NN      50000   // nodes
#define EE      800000  // edges per relation
#define DD      128     // feature dim (in == out)
#define MTILES  (NN / 16)   // 3125
#define NTILES  (DD / 16)   // 8

// ---------------------------------------------------------------------------
// Zero-init d_out and the two count arrays (graph-capture safe, no memset).
// ---------------------------------------------------------------------------
__global__ void hrgcn_zero(float* __restrict__ out, unsigned* __restrict__ cnt,
                           int nout, int ncnt) {
  int t = blockIdx.x * blockDim.x + threadIdx.x;
  if (t < nout) out[t] = 0.0f;
  if (t < ncnt) cnt[t] = 0u;
}

// ---------------------------------------------------------------------------
// Wh{r} = feat @ W{r} + b{r} using V_WMMA_F32_16X16X4_F32.
// One wave per 16x16 output tile; 3125*8*2 = 50000 waves exactly.
// A 16x4 f32 layout (ISA 7.12.2): lanes 0-15 -> K=0,1 ; lanes 16-31 -> K=2,3.
// B 4x16 f32: N striped across lanes, K across the 2 VGPRs (mirror of A).
// C/D 16x16 f32: VGPR v holds M = v + 8*(lane>>4), N = lane&15.
// ---------------------------------------------------------------------------
__global__ void hrgcn_gemm_wmma(const float* __restrict__ feat,
                                const float* __restrict__ W0,
                                const float* __restrict__ b0,
                                const float* __restrict__ W1,
                                const float* __restrict__ b1,
                                float* __restrict__ Wh0,
                                float* __restrict__ Wh1) {
  int wave = (blockIdx.x * blockDim.x + threadIdx.x) >> 5;
  int lane = threadIdx.x & 31;

  int rel = wave / (MTILES * NTILES);
  int rem = wave % (MTILES * NTILES);
  int mt  = rem / NTILES;
  int nt  = rem % NTILES;

  const float* __restrict__ W    = rel ? W1 : W0;
  const float* __restrict__ bias = rel ? b1 : b0;
  float*       __restrict__ Wh   = rel ? Wh1 : Wh0;

  int row = lane & 15;     // M within tile (A) / N within tile (B,C,D)
  int kh  = lane >> 4;     // which K-pair half this lane holds

  const float* arow = feat + (size_t)(mt * 16 + row) * DD;
  const float* bcol = W + nt * 16 + row;   // row here = N for B operand

  v8f c = {};
#pragma unroll
  for (int k0 = 0; k0 < DD; k0 += 4) {
    int ka = k0 + 2 * kh;               // this lane's first K
    v2f a, b;
    a.x = arow[ka];
    a.y = arow[ka + 1];
    b.x = bcol[(size_t)ka * DD];
    b.y = bcol[(size_t)(ka + 1) * DD];
    // (neg_a, A, neg_b, B, c_mod, C, reuse_a, reuse_b)
    c = __builtin_amdgcn_wmma_f32_16x16x4_f32(false, a, false, b,
                                              (short)0, c, false, false);
  }

  float bv = bias[nt * 16 + row];
#pragma unroll
  for (int v = 0; v < 8; ++v) {
    int m = mt * 16 + v + 8 * kh;
    Wh[(size_t)m * DD + nt * 16 + row] = c[v] + bv;
  }
}

// ---------------------------------------------------------------------------
// Per-destination in-degree counts (both relations in one launch).
// ---------------------------------------------------------------------------
__global__ void hrgcn_count(const int* __restrict__ dst0,
                            const int* __restrict__ dst1,
                            unsigned* __restrict__ cnt0,
                            unsigned* __restrict__ cnt1) {
  int t = blockIdx.x * blockDim.x + threadIdx.x;
  if (t < EE) {
    atomicAdd(&cnt0[dst0[t]], 1u);
  } else if (t < 2 * EE) {
    atomicAdd(&cnt1[dst1[t - EE]], 1u);
  }
}

// ---------------------------------------------------------------------------
// Edge scatter: one wave per edge. Each lane moves float4 (128 feats / 32
// lanes), pre-scaled by 1/cnt[dst] so the atomic accumulation directly
// produces sum-of-means. Output nodes with no in-edges stay 0 (== max(cnt,1)
// semantics of the reference).
// ---------------------------------------------------------------------------
__global__ void hrgcn_scatter(const float* __restrict__ Wh0,
                              const float* __restrict__ Wh1,
                              const int* __restrict__ src0,
                              const int* __restrict__ dst0,
                              const int* __restrict__ src1,
                              const int* __restrict__ dst1,
                              const unsigned* __restrict__ cnt0,
                              const unsigned* __restrict__ cnt1,
                              float* __restrict__ out) {
  int gw   = (blockIdx.x * blockDim.x + threadIdx.x) >> 5;
  int lane = threadIdx.x & 31;
  if (gw >= 2 * EE) return;

  int rel = (gw >= EE) ? 1 : 0;
  int e   = rel ? (gw - EE) : gw;

  const float* __restrict__ Wh = rel ? Wh1 : Wh0;
  int s = rel ? src1[e] : src0[e];
  int d = rel ? dst1[e] : dst0[e];

  const float* wrow = Wh + (size_t)s * DD;
  __builtin_prefetch(wrow + lane * 4, 0, 0);   // global_prefetch on gfx1250

  unsigned c = rel ? cnt1[d] : cnt0[d];        // >= 1: this edge exists
  float scale = 1.0f / (float)c;

  const float4 v = *((const float4*)wrow + lane);
  float* o = out + (size_t)d * DD + lane * 4;
  atomicAdd(o + 0, v.x * scale);
  atomicAdd(o + 1, v.y * scale);
  atomicAdd(o + 2, v.z * scale);
  atomicAdd(o + 3, v.w * scale);
}

// ---------------------------------------------------------------------------
extern "C" void kernel_launch(void* const* d_in, const int* in_sizes, int n_in,
                              void* d_out, int out_size, void* d_ws, size_t ws_size,
                              hipStream_t stream) {
  const float* feat = (const float*)d_in[0];
  const float* W0   = (const float*)d_in[1];
  const float* b0   = (const float*)d_in[2];
  const float* W1   = (const float*)d_in[3];
  const float* b1   = (const float*)d_in[4];
  const int*   src0 = (const int*)d_in[5];
  const int*   dst0 = (const int*)d_in[6];
  const int*   src1 = (const int*)d_in[7];
  const int*   dst1 = (const int*)d_in[8];
  float* out = (float*)d_out;

  // Workspace layout: Wh0 | Wh1 | cnt0 | cnt1
  float*    Wh0  = (float*)d_ws;
  float*    Wh1  = Wh0 + (size_t)NN * DD;
  unsigned* cnt0 = (unsigned*)(Wh1 + (size_t)NN * DD);
  unsigned* cnt1 = cnt0 + NN;

  // 1) zero output + counts
  {
    int nout = NN * DD;            // 6.4M
    int ncnt = 2 * NN;             // cnt0,cnt1 contiguous
    int blocks = (nout + 255) / 256;
    hrgcn_zero<<<blocks, 256, 0, stream>>>(out, cnt0, nout, ncnt);
  }

  // 2) WMMA GEMMs: 50000 waves exactly -> 6250 blocks of 8 waves
  hrgcn_gemm_wmma<<<(MTILES * NTILES * 2) / 8, 256, 0, stream>>>(
      feat, W0, b0, W1, b1, Wh0, Wh1);

  // 3) in-degree counts: 1.6M threads
  hrgcn_count<<<(2 * EE + 255) / 256, 256, 0, stream>>>(dst0, dst1, cnt0, cnt1);

  // 4) edge scatter: one wave per edge, 1.6M waves -> 200000 blocks
  hrgcn_scatter<<<(2 * EE) / 8, 256, 0, stream>>>(
      Wh0, Wh1, src0, dst0, src1, dst1, cnt0, cnt1, out);
}